// DifferentiableRoIAlignRotated_18107582120057
// MI455X (gfx1250) — compile-verified
//
#include <hip/hip_runtime.h>
#include <cstdint>

#define OUT_HW 49          // 7x7
#define CCH    256
#define FH     128
#define FW     128
#define FHW    (FH*FW)
#define FCHW   (CCH*FHW)
#define SCALE  0.0625f

struct __align__(16) PtInfo {
    int   off[4];   // clamped spatial offsets (y*W + x) for the 4 corners
    float w[4];     // bilinear weights with OOB mask folded in (0 if invalid)
};

// Issue 4 per-lane async gathers (global -> this lane's LDS slots).
// ASYNCcnt +4 per wave; completion is in issue order, so s_wait_asynccnt N
// fences exactly the oldest (cnt-N) of them.
__device__ __forceinline__ void async_gather4(unsigned lds,
                                              const float* __restrict__ cb,
                                              const PtInfo& pi)
{
    asm volatile("global_load_async_to_lds_b32 %0, %1, off"
                 :: "v"(lds),      "v"(cb + pi.off[0]) : "memory");
    asm volatile("global_load_async_to_lds_b32 %0, %1, off"
                 :: "v"(lds + 4u), "v"(cb + pi.off[1]) : "memory");
    asm volatile("global_load_async_to_lds_b32 %0, %1, off"
                 :: "v"(lds + 8u), "v"(cb + pi.off[2]) : "memory");
    asm volatile("global_load_async_to_lds_b32 %0, %1, off"
                 :: "v"(lds + 12u), "v"(cb + pi.off[3]) : "memory");
}

__global__ __launch_bounds__(256, 4)
void roialign_rotated_kernel(const float* __restrict__ feat,
                             const float* __restrict__ rois,
                             float* __restrict__ out)
{
    const int k   = blockIdx.x;
    const int tid = threadIdx.x;

    __shared__ PtInfo pts[OUT_HW];
    __shared__ float  gbuf[2][256 * 4];   // double-buffered async gather slots
    __shared__ int    s_fb;

    // ---- per-ROI geometry (49 threads) --------------------------------
    if (tid < OUT_HW) {
        const float* r = rois + k * 6;       // [batch, cx, cy, w, h, theta]
        const float cx = r[1] * SCALE;
        const float cy = r[2] * SCALE;
        const float rw = r[3] * SCALE;
        const float rh = r[4] * SCALE;
        const float th = r[5] * SCALE;       // reference scales theta too
        if (tid == 0) s_fb = (int)r[0] * FCHW;

        const int   py  = tid / 7;
        const int   px  = tid - py * 7;
        const float bgx = (float)px * (1.0f / 6.0f) - 0.5f;
        const float bgy = (float)py * (1.0f / 6.0f) - 0.5f;
        const float ct  = cosf(th);
        const float st  = sinf(th);

        const float gx = bgx * rw;
        const float gy = bgy * rh;
        const float xs = gx * ct - gy * st + cx;
        const float ys = gx * st + gy * ct + cy;

        const float xg = 2.0f * xs / 127.0f - 1.0f;
        const float yg = 2.0f * ys / 127.0f - 1.0f;
        const float ix = ((xg + 1.0f) * 128.0f - 1.0f) * 0.5f;
        const float iy = ((yg + 1.0f) * 128.0f - 1.0f) * 0.5f;

        const float x0 = floorf(ix), y0 = floorf(iy);
        const float wx1 = ix - x0,  wy1 = iy - y0;
        const float wx0 = 1.0f - wx1, wy0 = 1.0f - wy1;

        const float xf[2]  = { x0, x0 + 1.0f };
        const float yf[2]  = { y0, y0 + 1.0f };
        const float wxa[2] = { wx0, wx1 };
        const float wya[2] = { wy0, wy1 };

        PtInfo pi;
#pragma unroll
        for (int jy = 0; jy < 2; ++jy) {
#pragma unroll
            for (int jx = 0; jx < 2; ++jx) {
                const int   j  = jy * 2 + jx;
                const float xx = xf[jx], yy = yf[jy];
                const bool valid = (xx >= 0.0f) && (xx <= 127.0f) &&
                                   (yy >= 0.0f) && (yy <= 127.0f);
                const int xc = (int)fminf(fmaxf(xx, 0.0f), 127.0f);
                const int yc = (int)fminf(fmaxf(yy, 0.0f), 127.0f);
                pi.off[j] = yc * FW + xc;
                pi.w[j]   = valid ? (wya[jy] * wxa[jx]) : 0.0f;
            }
        }
        pts[tid] = pi;
    }
    __syncthreads();

    const float* fb = feat + s_fb;
    float*       ob = out + (size_t)k * (CCH * OUT_HW);

    // Per-lane LDS byte addresses of the two gather buffers (low 32 bits of a
    // generic pointer into LDS == wave-relative LDS byte offset).
    const unsigned lb0 = (unsigned)(uintptr_t)(&gbuf[0][tid * 4]);
    const unsigned lb1 = (unsigned)(uintptr_t)(&gbuf[1][tid * 4]);

    // ---- software-pipelined async gather loop -------------------------
    // 49 iterations * 256 threads == 12544 == C*49; idx == flat (c*49 + p)
    // output index, so the NT stores are fully contiguous across the block.

    // Prologue: stage 0 -> buffer 0.
    {
        const int c0 = tid / 49;
        const int p0 = tid - c0 * 49;
        PtInfo cur = pts[p0];
        async_gather4(lb0, fb + c0 * FHW, cur);

        // Keep stage state in registers across iterations.
        PtInfo stage = cur;

#pragma unroll 2
        for (int it = 0; it < 48; ++it) {
            // Issue stage it+1 into the other buffer.
            const int nidx = (it + 1) * 256 + tid;
            const int nc   = nidx / 49;
            const int np   = nidx - nc * 49;
            const PtInfo nxt = pts[np];
            const unsigned nl = ((it + 1) & 1) ? lb1 : lb0;
            async_gather4(nl, fb + nc * FHW, nxt);

            // Oldest 4 gathers (stage it) done; newest 4 still in flight.
            asm volatile("s_wait_asynccnt 0x4" ::: "memory");

            const float* s = (it & 1) ? &gbuf[1][tid * 4] : &gbuf[0][tid * 4];
            float acc =      s[0] * stage.w[0];
            acc = fmaf(s[1], stage.w[1], acc);
            acc = fmaf(s[2], stage.w[2], acc);
            acc = fmaf(s[3], stage.w[3], acc);

            // NT store: keep the 33.5 MB feature map resident in L2 while
            // 205 MB of output streams past.
            __builtin_nontemporal_store(acc, ob + it * 256 + tid);

            stage = nxt;
        }

        // Epilogue: stage 48 (buffer 0).
        asm volatile("s_wait_asynccnt 0x0" ::: "memory");
        const float* s = &gbuf[0][tid * 4];
        float acc =      s[0] * stage.w[0];
        acc = fmaf(s[1], stage.w[1], acc);
        acc = fmaf(s[2], stage.w[2], acc);
        acc = fmaf(s[3], stage.w[3], acc);
        __builtin_nontemporal_store(acc, ob + 48 * 256 + tid);
    }
}

extern "C" void kernel_launch(void* const* d_in, const int* in_sizes, int n_in,
                              void* d_out, int out_size, void* d_ws, size_t ws_size,
                              hipStream_t stream)
{
    const float* feat = (const float*)d_in[0];   // (2,256,128,128) f32
    const float* rois = (const float*)d_in[1];   // (K,6) f32
    float*       out  = (float*)d_out;           // (K,256,7,7) f32

    const int K = in_sizes[1] / 6;
    roialign_rotated_kernel<<<K, 256, 0, stream>>>(feat, rois, out);
}